// SAGraphLayerV3_21500606284195
// MI455X (gfx1250) — compile-verified
//
#include <hip/hip_runtime.h>

#define D 128

typedef float v2f __attribute__((ext_vector_type(2)));
typedef float v8f __attribute__((ext_vector_type(8)));

// ---------------- kernel 1: zero workspace (agg + deg) ----------------
__global__ void zero_kernel(float* __restrict__ p, long long n) {
    long long i = (long long)blockIdx.x * blockDim.x + threadIdx.x;
    long long stride = (long long)gridDim.x * blockDim.x;
    for (; i < n; i += stride) p[i] = 0.0f;
}

// ---------------- kernel 2: edge scatter of RAW features ----------------
// agg[dst] += x[src]; deg[dst] += 1.  One wave32 per edge, 4 floats per lane.
__device__ __forceinline__ void fatomic_add(float* p, float v) {
    unsafeAtomicAdd(p, v);   // lowers to global_atomic_add_f32 on gfx1250
}

__global__ void __launch_bounds__(256)
edge_scatter_kernel(const float* __restrict__ x,
                    const long long* __restrict__ ei,   // (2, E) int64
                    float* __restrict__ agg,
                    float* __restrict__ deg,
                    int E) {
    int gtid = blockIdx.x * blockDim.x + threadIdx.x;
    int e    = gtid >> 5;                 // one wave32 per edge
    int lane = threadIdx.x & 31;
    if (e >= E) return;                   // wave-uniform branch

    long long s = ei[e];                  // src row
    long long d = ei[(long long)E + e];   // dst row

    const float4 v = *(const float4*)(x + s * (long long)D + lane * 4);
    float* a = agg + d * (long long)D + lane * 4;
    fatomic_add(a + 0, v.x);
    fatomic_add(a + 1, v.y);
    fatomic_add(a + 2, v.z);
    fatomic_add(a + 3, v.w);
    if (lane == 0) fatomic_add(deg + d, 1.0f);
}

// ---------------- kernel 3: fused dual GEMM via V_WMMA_F32_16X16X4_F32 ----------
// out[m,:] = x[m,:] @ Wself + ((agg[m,:] - deg*x[m,:])/max(deg,1)) @ Wneigh + bias
// One wave per 16-row tile; 8 waves (256 threads) per block.
__global__ void __launch_bounds__(256)
node_gemm_kernel(const float* __restrict__ x,
                 const float* __restrict__ agg,
                 const float* __restrict__ deg,
                 const float* __restrict__ Wself,    // (K=128, N=128) row-major
                 const float* __restrict__ Wneigh,   // (K=128, N=128) row-major
                 const float* __restrict__ bias,
                 float* __restrict__ out,
                 int ntiles) {
    int wave = threadIdx.x >> 5;
    int lane = threadIdx.x & 31;
    int tile = blockIdx.x * 8 + wave;
    if (tile >= ntiles) return;           // wave-uniform; EXEC stays all-1s inside

    int row_base = tile * 16;
    int m     = lane & 15;                // row within tile (A) / column within n-tile (B,C,D)
    int hi    = lane >> 4;                // half-wave select
    int khalf = hi * 2;                   // A/B layout: lanes16-31 carry K+2,K+3

    const float* xrow = x   + (size_t)(row_base + m) * D;
    const float* grow = agg + (size_t)(row_base + m) * D;
    float dg = deg[row_base + m];
    float rd = 1.0f / fmaxf(dg, 1.0f);

    v8f acc[8];
    v8f zero = {};
#pragma unroll
    for (int t = 0; t < 8; ++t) acc[t] = zero;

    for (int k0 = 0; k0 < D; k0 += 4) {
        // A fragment (16x4 f32): lane<16 -> A[m][k0+0..1], lane>=16 -> A[m][k0+2..3]
        v2f a_x = *(const v2f*)(xrow + k0 + khalf);
        v2f a_g = *(const v2f*)(grow + k0 + khalf);
        v2f a_h;
        a_h.x = (a_g.x - dg * a_x.x) * rd;
        a_h.y = (a_g.y - dg * a_x.y) * rd;

        const float* ws = Wself  + (size_t)(k0 + khalf) * D + m;
        const float* wn = Wneigh + (size_t)(k0 + khalf) * D + m;

#pragma unroll
        for (int t = 0; t < 8; ++t) {
            int n0 = t * 16;
            // B fragment (4x16 f32): VGPR0 = W[krow][n], VGPR1 = W[krow+1][n]
            v2f bs; bs.x = ws[n0]; bs.y = ws[n0 + D];
            v2f bn; bn.x = wn[n0]; bn.y = wn[n0 + D];

            acc[t] = __builtin_amdgcn_wmma_f32_16x16x4_f32(
                         false, a_x, false, bs, (short)0, acc[t], false, false);
            acc[t] = __builtin_amdgcn_wmma_f32_16x16x4_f32(
                         false, a_h, false, bn, (short)0, acc[t], false, false);
        }
    }

    // C/D layout: VGPR r -> row M = r + hi*8, column N = (lane&15) within n-tile
#pragma unroll
    for (int t = 0; t < 8; ++t) {
        int n0 = t * 16;
        float bz = bias[n0 + m];
        float* op = out + (size_t)(row_base + hi * 8) * D + n0 + m;
#pragma unroll
        for (int r = 0; r < 8; ++r) {
            op[(size_t)r * D] = acc[t][r] + bz;
        }
    }
}

// ---------------- launch ----------------
extern "C" void kernel_launch(void* const* d_in, const int* in_sizes, int n_in,
                              void* d_out, int out_size, void* d_ws, size_t ws_size,
                              hipStream_t stream) {
    const float*     x      = (const float*)d_in[0];
    const long long* ei     = (const long long*)d_in[1];   // int64 (2,E)
    const float*     Wself  = (const float*)d_in[2];
    const float*     Wneigh = (const float*)d_in[3];
    const float*     bias   = (const float*)d_in[4];
    float*           out    = (float*)d_out;

    int N = in_sizes[0] / D;      // 100000
    int E = in_sizes[1] / 2;      // 1600000

    float* agg = (float*)d_ws;                    // N*D floats
    float* deg = agg + (size_t)N * D;             // N floats

    long long zn = (long long)N * D + N;
    zero_kernel<<<2048, 256, 0, stream>>>(agg, zn);

    int edges_per_block = 256 / 32;               // 8 waves -> 8 edges per block
    int grid2 = (E + edges_per_block - 1) / edges_per_block;
    edge_scatter_kernel<<<grid2, 256, 0, stream>>>(x, ei, agg, deg, E);

    int ntiles = (N + 15) / 16;                   // 6250, exact
    int grid3  = (ntiles + 7) / 8;
    node_gemm_kernel<<<grid3, 256, 0, stream>>>(x, agg, deg, Wself, Wneigh,
                                                bias, out, ntiles);
}